// learned_qp_solver_27187142984036
// MI455X (gfx1250) — compile-verified
//
#include <hip/hip_runtime.h>
#include <math.h>

typedef __attribute__((ext_vector_type(2))) float v2f;
typedef __attribute__((ext_vector_type(8))) float v8f;

constexpr int kB    = 4096;   // batch
constexpr int kIND  = 260;    // input dim (K of GEMM1, multiple of 4)
constexpr int kH    = 1024;   // hidden
constexpr int kN    = 256;    // NUM
constexpr int kNC   = 1530;   // constraints
constexpr int kSP   = 1536;   // padded NC
constexpr int kOUTR = 1786;   // raw MLP out
constexpr int kOUTP = 1792;   // padded MLP out (mult of 128)
constexpr int kKK   = 260;    // padded KKT K (257 -> 260, mult of 4)
constexpr int kKN   = 384;    // padded KKT N (257 -> 384, mult of 128)
constexpr int kIT   = 5;      // MAXITER

// ---- workspace layout (float offsets) ----
constexpr size_t OFF_H    = 0;
constexpr size_t OFF_W2P  = OFF_H    + (size_t)kB * kH;
constexpr size_t OFF_B2P  = OFF_W2P  + (size_t)kH * kOUTP;
constexpr size_t OFF_NOUT = OFF_B2P  + (size_t)kOUTP;
constexpr size_t OFF_LAM  = OFF_NOUT + (size_t)kB * kOUTP;
constexpr size_t OFF_S    = OFF_LAM  + (size_t)kB * kN;
constexpr size_t OFF_RHS  = OFF_S    + (size_t)kB * kSP;
constexpr size_t OFF_SOL  = OFF_RHS  + (size_t)kB * kKK;
constexpr size_t OFF_BINV = OFF_SOL  + (size_t)kB * kKN;
constexpr size_t OFF_BCON = OFF_BINV + (size_t)kKK * kKN;
constexpr size_t OFF_ZW   = OFF_BCON + (size_t)kSP;       // 255 x 288 solver scratch

// Q = 3I + 800*D^T D + 320000*D2^T D2  (exact fp32 integer entries)
__device__ __forceinline__ float costQ(int i, int j) {
  int d = i - j; d = d < 0 ? -d : d;
  if (d == 0) {
    float dtd = (i == 0 || i == 255) ? 1.f : 2.f;
    float d2  = (i == 0 || i == 255) ? 1.f : ((i == 1 || i == 254) ? 5.f : 6.f);
    return 3.f + 800.f * dtd + 320000.f * d2;
  } else if (d == 1) {
    int m = i < j ? i : j;
    float d2 = (m == 0 || m == 254) ? -2.f : -4.f;
    return -800.f + 320000.f * d2;
  } else if (d == 2) {
    return 320000.f;
  }
  return 0.f;
}

// g[j] = (u @ A_control)[j] using A_control structure; u is length-1536 (pad=0)
__device__ __forceinline__ float dotA(const float* u, int j) {
  float g = u[j] - u[256 + j];
  float wj  = (j <= 254) ? (u[512 + j]     - u[767 + j])     : 0.f;
  float wm1 = (j >= 1)   ? (u[512 + j - 1] - u[767 + j - 1]) : 0.f;
  g += 20.f * (wm1 - wj);
  float vj  = (j <= 253)           ? (u[1022 + j]     - u[1276 + j])     : 0.f;
  float vm1 = (j >= 1 && j <= 254) ? (u[1022 + j - 1] - u[1276 + j - 1]) : 0.f;
  float vm2 = (j >= 2)             ? (u[1022 + j - 2] - u[1276 + j - 2]) : 0.f;
  g += 400.f * (vj - 2.f * vm1 + vm2);
  return g;
}

__device__ __forceinline__ float blockReduce256(float v, float* red) {
  int t = threadIdx.x;
  red[t] = v; __syncthreads();
  for (int off = 128; off > 0; off >>= 1) {
    if (t < off) red[t] += red[t + off];
    __syncthreads();
  }
  float r = red[0];
  __syncthreads();
  return r;
}

// ------------------- WMMA f32 GEMM: C = A(MxK) * B(KxN) [+bias][ReLU] -------------------
// block = 8 waves -> 128x128 tile, wave -> 32x64 tile, V_WMMA_F32_16X16X4_F32
// 8 WMMAs per k-step (2 A-frags x 4 B-frags), running B pointer (no per-iter 64b mul),
// speculative global_prefetch on the B stream.
template<int EPI>  // 0: none, 1: +bias, 2: relu(+bias)
__global__ __launch_bounds__(256) void k_gemm(const float* __restrict__ A,
                                              const float* __restrict__ Bm,
                                              const float* __restrict__ bias,
                                              float* __restrict__ C,
                                              int K, int lda, int ldb, int ldc) {
  const int lane = threadIdx.x & 31;
  const int wave = threadIdx.x >> 5;
  const int wm   = wave >> 1;            // 0..3 (M)
  const int wn   = wave & 1;             // 0..1 (N)
  const int half = lane >> 4;            // 0/1
  const int l    = lane & 15;
  const int m0   = blockIdx.x * 128 + wm * 32;
  const int n0   = blockIdx.y * 128 + wn * 64;

  v8f acc[2][4] = {};

  const float* ap0 = A + (size_t)(m0 + l) * lda + 2 * half;        // A[m0+l][2h + k]
  const float* ap1 = A + (size_t)(m0 + 16 + l) * lda + 2 * half;
  const float* bp  = Bm + (size_t)(2 * half) * ldb + n0 + l;       // B[2h + k][n0+l]

  for (int k = 0; k < K; k += 4) {
    v2f a0 = *(const v2f*)(ap0 + k);
    v2f a1 = *(const v2f*)(ap1 + k);
    v2f b[4];
#pragma unroll
    for (int ni = 0; ni < 4; ++ni) {
      b[ni].x = bp[16 * ni];
      b[ni].y = bp[(size_t)ldb + 16 * ni];
    }
    __builtin_prefetch(bp + (size_t)16 * ldb, 0, 1);               // global_prefetch_b8
#pragma unroll
    for (int ni = 0; ni < 4; ++ni) {
      acc[0][ni] = __builtin_amdgcn_wmma_f32_16x16x4_f32(false, a0, false, b[ni], (short)0, acc[0][ni], false, false);
      acc[1][ni] = __builtin_amdgcn_wmma_f32_16x16x4_f32(false, a1, false, b[ni], (short)0, acc[1][ni], false, false);
    }
    bp += (size_t)4 * ldb;
  }

  float biasv[4] = {0.f, 0.f, 0.f, 0.f};
  if (EPI >= 1) {
#pragma unroll
    for (int ni = 0; ni < 4; ++ni) biasv[ni] = bias[n0 + 16 * ni + l];
  }
#pragma unroll
  for (int mi = 0; mi < 2; ++mi) {
    const int rbase = m0 + 16 * mi + 8 * half;
#pragma unroll
    for (int r = 0; r < 8; ++r) {
      float* crow = C + (size_t)(rbase + r) * ldc + n0 + l;
#pragma unroll
      for (int ni = 0; ni < 4; ++ni) {
        float v = acc[mi][ni][r];
        if (EPI >= 1) v += biasv[ni];
        if (EPI == 2) v = fmaxf(v, 0.f);
        crow[16 * ni] = v;
      }
    }
  }
}

// ------------------- small setup kernels -------------------
__global__ __launch_bounds__(256) void k_bcontrol(float* bcon,
    const float* smax, const float* smin, const float* sdmax,
    const float* sdmin, const float* sddmax, const float* sddmin) {
  int c = blockIdx.x * 256 + threadIdx.x;      // 0..1535
  float v = 0.f;
  if      (c < 256)  v =  smax[0];
  else if (c < 512)  v = -smin[0];
  else if (c < 767)  v =  sdmax[0];
  else if (c < 1022) v = -sdmin[0];
  else if (c < 1276) v =  sddmax[0];
  else if (c < 1530) v = -sddmin[0];
  bcon[c] = v;
}

__global__ __launch_bounds__(256) void k_padW2(float* W2p, float* b2p,
                                               const float* W2, const float* b2) {
  size_t idx = (size_t)blockIdx.x * 256 + threadIdx.x;   // over kH*kOUTP
  int n = (int)(idx % kOUTP);
  int k = (int)(idx / kOUTP);
  W2p[idx] = (n < kOUTR) ? W2[(size_t)k * kOUTR + n] : 0.f;
  if (idx < (size_t)kOUTP) b2p[idx] = (idx < (size_t)kOUTR) ? b2[idx] : 0.f;
}

// Build Binv[c][n] = inv(cost_mat)[n][c] (row c = solution of cost_mat*y = e_c),
// via banded Cholesky of Q22 (255x255, bandwidth 2). One block, 288 threads.
__global__ __launch_bounds__(288) void k_inverse(float* __restrict__ Binv,
                                                 float* __restrict__ zw) {
  __shared__ float Ld[255], L1s[255], L2s[255];
  const int t = threadIdx.x;
  for (size_t i = t; i < (size_t)kKK * kKN; i += 288) Binv[i] = 0.f;  // zero pads
  if (t == 0) {
    for (int i = 0; i < 255; ++i) {
      float l2 = (i >= 2) ? costQ(i + 1, i - 1) / Ld[i - 2] : 0.f;
      float l1 = (i >= 1) ? (costQ(i + 1, i) - l2 * L1s[i - 1]) / Ld[i - 1] : 0.f;
      L2s[i] = l2; L1s[i] = l1;
      Ld[i] = sqrtf(costQ(i + 1, i + 1) - l1 * l1 - l2 * l2);
    }
  }
  __syncthreads();
  if (t < 257) {
    const int c = t;
    const float x0 = (c == 256) ? 1.f : 0.f;
    // forward: L z = r ; r[j] = e_c[j+1] - Q[j+1][0]*x0
    float zm1 = 0.f, zm2 = 0.f;
    for (int j = 0; j < 255; ++j) {
      float r = ((j + 1) == c) ? 1.f : 0.f;
      if (c == 256) {
        if (j == 0) r -= costQ(1, 0);
        if (j == 1) r -= costQ(2, 0);
      }
      float z = (r - L1s[j] * zm1 - L2s[j] * zm2) / Ld[j];
      zw[(size_t)j * 288 + c] = z;
      zm2 = zm1; zm1 = z;
    }
    // backward: L^T x = z (in place)
    float xp1 = 0.f, xp2 = 0.f;
    for (int i = 254; i >= 0; --i) {
      float l1n = (i + 1 <= 254) ? L1s[i + 1] : 0.f;
      float l2n = (i + 2 <= 254) ? L2s[i + 2] : 0.f;
      float x = (zw[(size_t)i * 288 + c] - l1n * xp1 - l2n * xp2) / Ld[i];
      zw[(size_t)i * 288 + c] = x;
      xp2 = xp1; xp1 = x;
    }
    float* Brow = Binv + (size_t)c * kKN;
    Brow[0] = x0;
    for (int j = 0; j < 255; ++j) Brow[j + 1] = zw[(size_t)j * 288 + c];
    float y1 = Brow[1], y2 = Brow[2];
    float nu = ((c == 0) ? 1.f : 0.f) - (costQ(0, 0) * x0 + costQ(0, 1) * y1 + costQ(0, 2) * y2);
    Brow[256] = nu;
  }
}

// lamda = nout[:, :256]; s = relu(nout[:, 256:1786]) padded to 1536
__global__ __launch_bounds__(256) void k_init(const float* __restrict__ nout,
                                              float* __restrict__ lam,
                                              float* __restrict__ s) {
  const int b = blockIdx.x, t = threadIdx.x;
  const float* nr = nout + (size_t)b * kOUTP;
  lam[(size_t)b * kN + t] = nr[t];
  float* sr = s + (size_t)b * kSP;
  for (int q = 0; q < 6; ++q) {
    int c = t + 256 * q;
    sr[c] = (c < kNC) ? fmaxf(nr[kN + c], 0.f) : 0.f;
  }
}

// rhs[b][:256] = lamda + steer_samples + (b_control - s) @ A_control ; rhs[b][256] = steer_init
__global__ __launch_bounds__(256) void k_rhs(const float* __restrict__ s,
                                             const float* __restrict__ lam,
                                             const float* __restrict__ ssamp,
                                             const float* __restrict__ sinit,
                                             const float* __restrict__ bcon,
                                             float* __restrict__ rhs) {
  __shared__ float u[kSP];
  const int b = blockIdx.x, t = threadIdx.x;
  const float* sr = s + (size_t)b * kSP;
  for (int q = 0; q < 6; ++q) {
    int c = t + 256 * q;
    u[c] = bcon[c] - sr[c];                 // pad entries: 0 - 0 = 0
  }
  __syncthreads();
  float g = dotA(u, t);
  float* rr = rhs + (size_t)b * kKK;
  rr[t] = lam[(size_t)b * kN + t] + ssamp[(size_t)b * kN + t] + g;
  if (t == 0) { rr[256] = sinit[b]; rr[257] = 0.f; rr[258] = 0.f; rr[259] = 0.f; }
}

// per-iteration update: As, s_new, residual, lamda update, norms, outputs
__global__ __launch_bounds__(256) void k_update(const float* __restrict__ sol,
                                                const float* __restrict__ bcon,
                                                float* __restrict__ s,
                                                float* __restrict__ lam,
                                                float* __restrict__ outProj,
                                                float* __restrict__ primStack,
                                                float* __restrict__ fpStack,
                                                int it) {
  __shared__ float xs[kN + 2];
  __shared__ float res[kSP];
  __shared__ float red[256];
  const int b = blockIdx.x, t = threadIdx.x;
  xs[t] = sol[(size_t)b * kKN + t];
  if (t < 2) xs[kN + t] = 0.f;
  __syncthreads();

  float r2 = 0.f, ds2 = 0.f;
  float* srow = s + (size_t)b * kSP;
  for (int q = 0; q < 6; ++q) {
    int c = t + 256 * q;
    if (c < kNC) {
      float As = 0.f;
      if      (c < 256)  As = xs[c];
      else if (c < 512)  As = -xs[c - 256];
      else if (c < 767)  { int i = c - 512;  As =  20.f * (xs[i + 1] - xs[i]); }
      else if (c < 1022) { int i = c - 767;  As = -20.f * (xs[i + 1] - xs[i]); }
      else if (c < 1276) { int i = c - 1022; As =  400.f * (xs[i] - 2.f * xs[i + 1] + xs[i + 2]); }
      else               { int i = c - 1276; As = -400.f * (xs[i] - 2.f * xs[i + 1] + xs[i + 2]); }
      float bc = bcon[c];
      float sn = fmaxf(bc - As, 0.f);
      float rv = fmaxf(As - bc, 0.f);      // == As - bc + sn exactly
      float dsv = srow[c] - sn;
      srow[c] = sn;
      res[c] = rv;
      r2 += rv * rv; ds2 += dsv * dsv;
    } else {
      res[c] = 0.f;
    }
  }
  float r2t  = blockReduce256(r2, red);    // barriers inside make res visible
  float ds2t = blockReduce256(ds2, red);

  float dl = dotA(res, t);
  float lv = lam[(size_t)b * kN + t] - dl;
  lam[(size_t)b * kN + t] = lv;
  float dl2t = blockReduce256(dl * dl, red);

  if (t == 0) {
    primStack[(size_t)it * kB + b] = sqrtf(r2t);
    fpStack[(size_t)it * kB + b]   = sqrtf(ds2t) + sqrtf(dl2t);
  }
  if (it == kIT - 1) outProj[(size_t)b * kN + t] = xs[t];
}

__global__ __launch_bounds__(256) void k_acc(const float* __restrict__ primStack,
                                             const float* __restrict__ fpStack,
                                             float* __restrict__ accP,
                                             float* __restrict__ accF) {
  int b = blockIdx.x * 256 + threadIdx.x;
  float p = 0.f, f = 0.f;
  for (int it = 0; it < kIT; ++it) {
    p += primStack[(size_t)it * kB + b];
    f += fpStack[(size_t)it * kB + b];
  }
  accP[b] = p / 5.f;
  accF[b] = f / 5.f;
}

extern "C" void kernel_launch(void* const* d_in, const int* in_sizes, int n_in,
                              void* d_out, int out_size, void* d_ws, size_t ws_size,
                              hipStream_t stream) {
  (void)in_sizes; (void)n_in; (void)out_size; (void)ws_size;
  const float* inp    = (const float*)d_in[0];
  const float* sinit  = (const float*)d_in[1];
  const float* ssamp  = (const float*)d_in[2];
  const float* smax   = (const float*)d_in[3];
  const float* smin   = (const float*)d_in[4];
  const float* sdmax  = (const float*)d_in[5];
  const float* sdmin  = (const float*)d_in[6];
  const float* sddmax = (const float*)d_in[7];
  const float* sddmin = (const float*)d_in[8];
  const float* W1     = (const float*)d_in[11];
  const float* b1     = (const float*)d_in[12];
  const float* W2     = (const float*)d_in[13];
  const float* b2     = (const float*)d_in[14];

  float* ws = (float*)d_ws;
  float* h    = ws + OFF_H;
  float* W2p  = ws + OFF_W2P;
  float* b2p  = ws + OFF_B2P;
  float* nout = ws + OFF_NOUT;
  float* lam  = ws + OFF_LAM;
  float* s    = ws + OFF_S;
  float* rhs  = ws + OFF_RHS;
  float* sol  = ws + OFF_SOL;
  float* Binv = ws + OFF_BINV;
  float* bcon = ws + OFF_BCON;
  float* zw   = ws + OFF_ZW;

  float* out       = (float*)d_out;
  float* outProj   = out;
  float* primStack = out + (size_t)kB * kN;
  float* fpStack   = primStack + (size_t)kIT * kB;
  float* accP      = fpStack + (size_t)kIT * kB;
  float* accF      = accP + kB;

  k_bcontrol<<<6, 256, 0, stream>>>(bcon, smax, smin, sdmax, sdmin, sddmax, sddmin);
  k_padW2<<<(kH * kOUTP) / 256, 256, 0, stream>>>(W2p, b2p, W2, b2);
  k_inverse<<<1, 288, 0, stream>>>(Binv, zw);

  // GEMM1: h = relu(inp @ W1 + b1)   (4096x260)*(260x1024)
  k_gemm<2><<<dim3(kB / 128, kH / 128), 256, 0, stream>>>(inp, W1, b1, h, kIND, kIND, kH, kH);
  // GEMM2: nout = h @ W2p + b2p      (4096x1024)*(1024x1792)
  k_gemm<1><<<dim3(kB / 128, kOUTP / 128), 256, 0, stream>>>(h, W2p, b2p, nout, kH, kH, kOUTP, kOUTP);

  k_init<<<kB, 256, 0, stream>>>(nout, lam, s);

  for (int it = 0; it < kIT; ++it) {
    k_rhs<<<kB, 256, 0, stream>>>(s, lam, ssamp, sinit, bcon, rhs);
    // GEMM3: sol = rhs @ Binv        (4096x260)*(260x384)
    k_gemm<0><<<dim3(kB / 128, kKN / 128), 256, 0, stream>>>(rhs, Binv, nullptr, sol, kKK, kKK, kKN, kKN);
    k_update<<<kB, 256, 0, stream>>>(sol, bcon, s, lam, outProj, primStack, fpStack, it);
  }
  k_acc<<<kB / 256, 256, 0, stream>>>(primStack, fpStack, accP, accF);
}